// GCN_80539226734706
// MI455X (gfx1250) — compile-verified
//
#include <hip/hip_runtime.h>
#include <cstdint>

typedef float v2f __attribute__((ext_vector_type(2)));
typedef float v8f __attribute__((ext_vector_type(8)));

#define BN_EPS 1e-5f

// ---------------- utility ----------------
__global__ __launch_bounds__(256) void fill_u32_kernel(uint32_t* __restrict__ p, size_t n, uint32_t v) {
    size_t i = (size_t)blockIdx.x * blockDim.x + threadIdx.x;
    if (i < n) p[i] = v;
}

// ---------------- graph normalization ----------------
__global__ __launch_bounds__(256) void deg_kernel(const float* __restrict__ ew,
                                                  const int* __restrict__ row,
                                                  float* __restrict__ deg, int E) {
    int e = blockIdx.x * blockDim.x + threadIdx.x;
    if (e < E) atomicAdd(&deg[row[e]], ew[e]);
}

__global__ __launch_bounds__(256) void dinv_kernel(float* __restrict__ deg, int n) {
    int i = blockIdx.x * blockDim.x + threadIdx.x;
    if (i < n) {
        float d = deg[i];
        deg[i] = (d > 0.0f) ? rsqrtf(d) : 0.0f;   // in-place: deg -> dinv
    }
}

__global__ __launch_bounds__(256) void norm_kernel(const float* __restrict__ ew,
                                                   const int* __restrict__ row,
                                                   const int* __restrict__ col,
                                                   const float* __restrict__ dinv,
                                                   float* __restrict__ norm, int E) {
    int e = blockIdx.x * blockDim.x + threadIdx.x;
    if (e < E) norm[e] = -ew[e] * dinv[row[e]] * dinv[col[e]];
}

// ---------------- sparse propagation: out[row] += norm * z[col] ----------------
__global__ __launch_bounds__(256) void prop_kernel(const float* __restrict__ z,
                                                   float* __restrict__ out,
                                                   const float* __restrict__ norm,
                                                   const int* __restrict__ row,
                                                   const int* __restrict__ col,
                                                   int E, int chunks, int fi) {
    size_t tid = (size_t)blockIdx.x * blockDim.x + threadIdx.x;
    size_t total = (size_t)E * chunks;
    if (tid >= total) return;
    int e = (int)(tid / chunks);
    int c = (int)(tid % chunks);
    float w = norm[e];
    int r  = row[e];
    int cl = col[e];
    const float4 v = *((const float4*)(z + (size_t)cl * fi) + c);
    float* o = out + (size_t)r * fi + (size_t)c * 4;
    atomicAdd(o + 0, w * v.x);
    atomicAdd(o + 1, w * v.y);
    atomicAdd(o + 2, w * v.z);
    atomicAdd(o + 3, w * v.w);
}

// T2 = 2*T2 - z
__global__ __launch_bounds__(256) void axpy_kernel(float* __restrict__ t2,
                                                   const float* __restrict__ z, size_t n) {
    size_t i = (size_t)blockIdx.x * blockDim.x + threadIdx.x;
    if (i < n) t2[i] = 2.0f * t2[i] - z[i];
}

// ---------------- fused Chebyshev GEMM: H = Z@W0 + T1@W1 + T2@W2 + b ----------------
// One wave computes a 16x16 fp32 tile with V_WMMA_F32_16X16X4_F32.
// Fragment layout (wave32): A (16x4): lanes 0-15 hold M=lane, K={0,1}; lanes 16-31 hold K={2,3}.
// B (4x16) mirrored: lanes 0-15 hold K={0,1} x N=lane; lanes 16-31 hold K={2,3}.
// C/D: VGPR r -> (M=r | r+8 for upper lanes), N = lane&15.
__global__ __launch_bounds__(256) void cheb_gemm_wmma(const float* __restrict__ Z,
                                                      const float* __restrict__ T1,
                                                      const float* __restrict__ T2,
                                                      const float* __restrict__ W,   // [3, fi, fo]
                                                      const float* __restrict__ bias,
                                                      float* __restrict__ H,
                                                      int n, int fi, int fo) {
    int wave = (blockIdx.x * blockDim.x + threadIdx.x) >> 5;
    int lane = threadIdx.x & 31;
    int tilesN = fo >> 4;
    int tilesM = n >> 4;                       // n is a multiple of 16 (50000 = 3125*16)
    if (wave >= tilesM * tilesN) return;       // wave-uniform: EXEC stays all-ones inside
    int tm = wave / tilesN;
    int tn = wave % tilesN;
    int half = lane >> 4;                      // 0: lanes 0-15, 1: lanes 16-31
    int l    = lane & 15;

    v8f acc = {};
    const float* srcs[3] = {Z, T1, T2};
    int arow = tm * 16 + l;
    int bcol = tn * 16 + l;
    for (int t = 0; t < 3; ++t) {
        const float* __restrict__ A = srcs[t];
        const float* __restrict__ B = W + (size_t)t * fi * fo;
        const float* arow_p = A + (size_t)arow * fi;
        for (int k = 0; k < fi; k += 4) {
            int ka = k + half * 2;
            v2f a, b;
            a.x = arow_p[ka];
            a.y = arow_p[ka + 1];
            b.x = B[(size_t)ka * fo + bcol];
            b.y = B[(size_t)(ka + 1) * fo + bcol];
            __builtin_prefetch(&B[(size_t)(ka + 4) * fo + bcol], 0, 1);
            acc = __builtin_amdgcn_wmma_f32_16x16x4_f32(
                false, a, false, b, (short)0, acc, false, false);
        }
    }
    float bv = bias[bcol];
#pragma unroll
    for (int r = 0; r < 8; ++r) {
        int rowi = tm * 16 + r + half * 8;
        H[(size_t)rowi * fo + bcol] = acc[r] + bv;
    }
}

// ---------------- softplus (in place) + per-feature sum/sumsq ----------------
__global__ __launch_bounds__(256) void softplus_stats_kernel(float* __restrict__ H,
                                                             float* __restrict__ stats, // [2*fo]
                                                             int n, int fo) {
    int t = blockIdx.x * blockDim.x + threadIdx.x;
    int f = t % fo;
    int r0 = (t / fo) * 8;
    if (r0 >= n) return;
    float s = 0.0f, sq = 0.0f;
    int r1 = min(r0 + 8, n);
    for (int r = r0; r < r1; ++r) {
        float v = H[(size_t)r * fo + f];
        float sp = fmaxf(v, 0.0f) + log1pf(expf(-fabsf(v)));  // stable softplus
        H[(size_t)r * fo + f] = sp;
        s += sp;
        sq += sp * sp;
    }
    atomicAdd(&stats[f], s);
    atomicAdd(&stats[fo + f], sq);
}

__global__ __launch_bounds__(256) void bn_apply_kernel(float* __restrict__ H,
                                                       const float* __restrict__ stats,
                                                       const float* __restrict__ g,
                                                       const float* __restrict__ be,
                                                       int n, int fo) {
    size_t i = (size_t)blockIdx.x * blockDim.x + threadIdx.x;
    if (i >= (size_t)n * fo) return;
    int f = (int)(i % fo);
    float inv_n = 1.0f / (float)n;
    float mean = stats[f] * inv_n;
    float var  = stats[fo + f] * inv_n - mean * mean;   // biased variance
    float scale = g[f] * rsqrtf(var + BN_EPS);
    H[i] = (H[i] - mean) * scale + be[f];
}

// ---------------- segment max / sum / count pooling ----------------
__global__ __launch_bounds__(256) void pool_kernel(const float* __restrict__ h,
                                                   const int* __restrict__ batch,
                                                   uint32_t* __restrict__ pmax,
                                                   float* __restrict__ psum,
                                                   int* __restrict__ pcnt,
                                                   int n, int fo) {
    size_t i = (size_t)blockIdx.x * blockDim.x + threadIdx.x;
    if (i >= (size_t)n * fo) return;
    int node = (int)(i / fo);
    int f    = (int)(i % fo);
    int g = batch[node];
    float v = h[i];
    uint32_t u = __float_as_uint(v);
    uint32_t ord = (u >> 31) ? ~u : (u | 0x80000000u);  // order-preserving map
    atomicMax(&pmax[(size_t)g * fo + f], ord);
    atomicAdd(&psum[(size_t)g * fo + f], v);
    if (f == 0) atomicAdd(&pcnt[g], 1);
}

// ---------------- dense head + log_softmax ----------------
__global__ __launch_bounds__(256) void head_kernel(const uint32_t* __restrict__ pmax,
                                                   const float* __restrict__ psum,
                                                   const int* __restrict__ pcnt,
                                                   const float* __restrict__ Wd,  // [512,4]
                                                   const float* __restrict__ bd,
                                                   float* __restrict__ out, int G, int fo) {
    __shared__ float logits[64][4];
    int t = threadIdx.x;
    int g = t >> 2, c = t & 3;
    if (g < G) {
        float acc = bd[c];
        float cnt = fmaxf((float)pcnt[g], 1.0f);
        for (int f = 0; f < fo; ++f) {
            uint32_t u = pmax[(size_t)g * fo + f];
            uint32_t bits = (u & 0x80000000u) ? (u & 0x7FFFFFFFu) : ~u;
            float mx = __uint_as_float(bits);
            if (!isfinite(mx)) mx = 0.0f;                 // empty-segment guard
            float mean = psum[(size_t)g * fo + f] / cnt;
            acc += mx * Wd[f * 4 + c] + mean * Wd[(fo + f) * 4 + c];
        }
        logits[g][c] = acc;
    }
    __syncthreads();
    if (g < G) {
        float m = fmaxf(fmaxf(logits[g][0], logits[g][1]),
                        fmaxf(logits[g][2], logits[g][3]));
        float s = 0.0f;
        for (int j = 0; j < 4; ++j) s += expf(logits[g][j] - m);
        out[g * 4 + c] = logits[g][c] - m - logf(s);
    }
}

// ---------------- host orchestration ----------------
extern "C" void kernel_launch(void* const* d_in, const int* in_sizes, int n_in,
                              void* d_out, int out_size, void* d_ws, size_t ws_size,
                              hipStream_t stream) {
    const int F = 128;
    const float* x     = (const float*)d_in[0];
    const float* ew    = (const float*)d_in[1];
    const int*   ei    = (const int*)d_in[2];
    const int*   batch = (const int*)d_in[3];
    const float* Ws[5] = {(const float*)d_in[4],  (const float*)d_in[8],
                          (const float*)d_in[12], (const float*)d_in[16],
                          (const float*)d_in[20]};
    const float* bs[5] = {(const float*)d_in[5],  (const float*)d_in[9],
                          (const float*)d_in[13], (const float*)d_in[17],
                          (const float*)d_in[21]};
    const float* gs[4]  = {(const float*)d_in[6],  (const float*)d_in[10],
                           (const float*)d_in[14], (const float*)d_in[18]};
    const float* bes[4] = {(const float*)d_in[7],  (const float*)d_in[11],
                           (const float*)d_in[15], (const float*)d_in[19]};
    const float* Wd = (const float*)d_in[22];
    const float* bd = (const float*)d_in[23];

    const int N = in_sizes[0] / F;      // 50000
    const int E = in_sizes[1];          // 1600000
    const int G = 64;
    const int* row = ei;
    const int* col = ei + E;

    // workspace carve (all 256B aligned)
    char* ws = (char*)d_ws;
    auto carve = [&](size_t bytes) -> void* {
        void* p = (void*)ws;
        ws += (bytes + 255) & ~(size_t)255;
        return p;
    };
    float*    norm  = (float*)carve((size_t)E * 4);
    float*    dinv  = (float*)carve((size_t)N * 4);            // deg then dinv
    float*    bufA  = (float*)carve((size_t)N * 256 * 4);
    float*    bufB  = (float*)carve((size_t)N * 256 * 4);
    float*    T1    = (float*)carve((size_t)N * 128 * 4);
    float*    T2    = (float*)carve((size_t)N * 128 * 4);
    float*    stats = (float*)carve(512 * 4);
    uint32_t* pmax  = (uint32_t*)carve((size_t)G * 256 * 4);
    float*    psum  = (float*)carve((size_t)G * 256 * 4);
    int*      pcnt  = (int*)carve((size_t)G * 4);

    auto blocks = [](size_t total) { return (unsigned)((total + 255) / 256); };
    auto zero = [&](void* p, size_t words) {
        fill_u32_kernel<<<blocks(words), 256, 0, stream>>>((uint32_t*)p, words, 0u);
    };

    // graph normalization
    zero(dinv, N);
    deg_kernel<<<blocks(E), 256, 0, stream>>>(ew, row, dinv, E);
    dinv_kernel<<<blocks(N), 256, 0, stream>>>(dinv, N);
    norm_kernel<<<blocks(E), 256, 0, stream>>>(ew, row, col, dinv, norm, E);

    const int dims[6] = {128, 16, 32, 64, 128, 256};
    const float* z = x;
    float* ping = bufA;
    float* pong = bufB;

    for (int l = 0; l < 5; ++l) {
        const int fi = dims[l], fo = dims[l + 1];
        const int chunks = fi / 4;
        const size_t nfi = (size_t)N * fi;
        const size_t nfo = (size_t)N * fo;

        // T1 = L_hat @ z
        zero(T1, nfi);
        prop_kernel<<<blocks((size_t)E * chunks), 256, 0, stream>>>(z, T1, norm, row, col, E, chunks, fi);
        // T2 = 2 * (L_hat @ T1) - z
        zero(T2, nfi);
        prop_kernel<<<blocks((size_t)E * chunks), 256, 0, stream>>>(T1, T2, norm, row, col, E, chunks, fi);
        axpy_kernel<<<blocks(nfi), 256, 0, stream>>>(T2, z, nfi);

        // H = z@W0 + T1@W1 + T2@W2 + b   (fp32 WMMA, one wave per 16x16 tile)
        float* H = ping;
        size_t tiles = (size_t)(N / 16) * (fo / 16);
        cheb_gemm_wmma<<<blocks(tiles * 32), 256, 0, stream>>>(z, T1, T2, Ws[l], bs[l], H, N, fi, fo);

        if (l < 4) {
            zero(stats, 2 * (size_t)fo);
            softplus_stats_kernel<<<blocks((size_t)((N + 7) / 8) * fo), 256, 0, stream>>>(H, stats, N, fo);
            bn_apply_kernel<<<blocks(nfo), 256, 0, stream>>>(H, stats, gs[l], bes[l], N, fo);
        }

        z = H;
        float* tmp = ping; ping = pong; pong = tmp;
    }

    // pooling over graphs (z now points at final [N,256] features)
    zero(pmax, (size_t)G * 256);
    zero(psum, (size_t)G * 256);
    zero(pcnt, G);
    pool_kernel<<<blocks((size_t)N * 256), 256, 0, stream>>>(z, batch, pmax, psum, pcnt, N, 256);

    // dense head + log_softmax -> d_out [G,4]
    head_kernel<<<1, 256, 0, stream>>>(pmax, psum, pcnt, Wd, bd, (float*)d_out, G, 256);
}